// Points_50156628082744
// MI455X (gfx1250) — compile-verified
//
#include <hip/hip_runtime.h>
#include <math.h>

typedef __attribute__((ext_vector_type(2))) float v2f;
typedef __attribute__((ext_vector_type(4))) float v4f;
typedef __attribute__((ext_vector_type(8))) float v8f;

#define CANVAS_H 256
#define CANVAS_W 384
#define NPTS     1024
#define R_WH     1.5f   // 384/256

// One wave handles one tile of 16 consecutive pixels against all 1024 points.
// Per 16-point chunk: 2x V_WMMA_F32_16X16X4_F32 compute the affine transform
//   t_j = gy*m0j + gx*m1j - b_j   for 16 pixels x 16 points,
// then elementwise |.|/relu + 3-channel color FMA in VALU on the C-layout.
//
// LDS layout is pre-swizzled per lane-half so the inner loop is branchless:
//   BJ0[(hi*N + n)] : {m00, m10} (lanes 0-15) | {-b0, 0} (lanes 16-31)
//   BJ1[(hi*N + n)] : {m01, m11}              | {-b1, 0}
//   C4[n]           : {r, g, b, 0}
__global__ __launch_bounds__(256) void points_kernel(
    const float* __restrict__ loc,   // (1,1,N,2)  [y,x]
    const float* __restrict__ mo,    // (N,2,2)
    const float* __restrict__ mse,   // (N,1,1)
    const float* __restrict__ col,   // (1,1,N,3)
    float* __restrict__ out)         // (3,H,W)
{
  __shared__ __align__(16) float lds[12 * NPTS];  // 48 KB
  float* BJ0 = lds;                 // 2*NPTS float2 = 4096 floats
  float* BJ1 = lds + 4 * NPTS;      // 2*NPTS float2 = 4096 floats
  float* C4  = lds + 8 * NPTS;      // NPTS float4   = 4096 floats

  // ---- per-point precompute into LDS (block-cooperative) ----
  for (int n = threadIdx.x; n < NPTS; n += blockDim.x) {
    float s   = 16.0f * __expf(-mse[n]);        // sqrt(1024)/2 / exp(e)
    float m00 = mo[n * 4 + 0] + s;
    float m01 = mo[n * 4 + 1];
    float m10 = mo[n * 4 + 2];
    float m11 = mo[n * 4 + 3] + s;
    float ly  = loc[n * 2 + 0];
    float lx  = loc[n * 2 + 1];
    float b0  = ly * m00 + lx * m10;
    float b1  = ly * m01 + lx * m11;

    v2f lo0; lo0.x = m00; lo0.y = m10;
    v2f hi0; hi0.x = -b0; hi0.y = 0.0f;
    v2f lo1; lo1.x = m01; lo1.y = m11;
    v2f hi1; hi1.x = -b1; hi1.y = 0.0f;
    *(v2f*)&BJ0[n * 2]              = lo0;
    *(v2f*)&BJ0[(NPTS + n) * 2]     = hi0;
    *(v2f*)&BJ1[n * 2]              = lo1;
    *(v2f*)&BJ1[(NPTS + n) * 2]     = hi1;
    v4f c; c.x = col[n * 3 + 0]; c.y = col[n * 3 + 1];
    c.z = col[n * 3 + 2]; c.w = 0.0f;
    *(v4f*)&C4[n * 4] = c;
  }
  __syncthreads();

  const int lane = threadIdx.x & 31;
  const int wave = threadIdx.x >> 5;
  const int tile = blockIdx.x * 8 + wave;       // 16 pixels per tile
  const int hi   = lane >> 4;                   // 0: lanes 0-15, 1: lanes 16-31
  const int ln   = lane & 15;

  // A-matrix (16x4 f32): lanes 0-15 hold K=0 (VGPR0) / K=1 (VGPR1) for M=lane,
  // lanes 16-31 hold K=2 / K=3. Rows = [gy, gx, 1, 0].
  int p0 = tile * 16 + lane;
  int h0 = p0 / CANVAS_W, w0 = p0 - h0 * CANVAS_W;
  float gy = -1.0f + (float)h0 * (2.0f / (float)(CANVAS_H - 1));
  float gx = -R_WH + (float)w0 * (2.0f * R_WH / (float)(CANVAS_W - 1));
  v2f a;
  a.x = hi ? 1.0f : gy;
  a.y = hi ? 0.0f : gx;

  float accR[8], accG[8], accB[8];
#pragma unroll
  for (int e = 0; e < 8; ++e) { accR[e] = 0.0f; accG[e] = 0.0f; accB[e] = 0.0f; }

  const v8f cz = {};
  const int bsel = (hi * NPTS + ln) * 2;        // lane-half base into BJ0/BJ1

  // ---- sweep all points in chunks of 16 columns (branchless feed) ----
  for (int k = 0; k < NPTS / 16; ++k) {
    v2f b0 = *(const v2f*)&BJ0[bsel + k * 32];
    v2f b1 = *(const v2f*)&BJ1[bsel + k * 32];
    v4f c4 = *(const v4f*)&C4[(k * 16 + ln) * 4];

    v8f d0 = __builtin_amdgcn_wmma_f32_16x16x4_f32(false, a, false, b0,
                                                   (short)0, cz, false, false);
    v8f d1 = __builtin_amdgcn_wmma_f32_16x16x4_f32(false, a, false, b1,
                                                   (short)0, cz, false, false);

    // C-layout element e on this lane = (pixel row e + 8*hi, point column ln).
#pragma unroll
    for (int e = 0; e < 8; ++e) {
      float dist = 0.5f * (fabsf(d0[e]) + fabsf(d1[e]));
      float m    = fmaxf(0.0f, 1.0f - dist);
      accR[e] = fmaf(m, c4.x, accR[e]);
      accG[e] = fmaf(m, c4.y, accG[e]);
      accB[e] = fmaf(m, c4.z, accB[e]);
    }
  }

  // ---- butterfly-reduce over the 16 lanes holding the same pixel row ----
#pragma unroll
  for (int e = 0; e < 8; ++e) {
#pragma unroll
    for (int off = 1; off < 16; off <<= 1) {
      accR[e] += __shfl_xor(accR[e], off, 16);
      accG[e] += __shfl_xor(accG[e], off, 16);
      accB[e] += __shfl_xor(accB[e], off, 16);
    }
  }

  // After the butterfly every lane in a 16-group holds the full sums for its
  // 8 rows; lane (ln==e) commits row e (+8 for the high half).
#pragma unroll
  for (int e = 0; e < 8; ++e) {
    if (ln == e) {
      int row = e + hi * 8;
      int p   = tile * 16 + row;
      int h   = p / CANVAS_W, w = p - h * CANVAS_W;
      float r = 1.0f / (1.0f + __expf(-4.0f * accR[e]));
      float g = 1.0f / (1.0f + __expf(-4.0f * accG[e]));
      float b = 1.0f / (1.0f + __expf(-4.0f * accB[e]));
      out[0 * CANVAS_H * CANVAS_W + h * CANVAS_W + w] = r;
      out[1 * CANVAS_H * CANVAS_W + h * CANVAS_W + w] = g;
      out[2 * CANVAS_H * CANVAS_W + h * CANVAS_W + w] = b;
    }
  }
}

extern "C" void kernel_launch(void* const* d_in, const int* in_sizes, int n_in,
                              void* d_out, int out_size, void* d_ws, size_t ws_size,
                              hipStream_t stream) {
  (void)in_sizes; (void)n_in; (void)out_size; (void)d_ws; (void)ws_size;
  const float* loc = (const float*)d_in[0];   // locations (1,1,N,2)
  const float* mo  = (const float*)d_in[1];   // matrix_offsets (N,2,2)
  const float* mse = (const float*)d_in[2];   // matrix_scale_exponents (N,1,1)
  const float* col = (const float*)d_in[3];   // colors (1,1,N,3)
  float* out = (float*)d_out;                 // (3,256,384)

  const int tiles  = (CANVAS_H * CANVAS_W) / 16;  // 6144
  const int blocks = tiles / 8;                    // 8 waves (tiles) per block
  points_kernel<<<blocks, 256, 0, stream>>>(loc, mo, mse, col, out);
}